// LSTM_54966991454470
// MI455X (gfx1250) — compile-verified
//
#include <hip/hip_runtime.h>

// Persistent single-workgroup 2-layer LSTM, T=65536 serial steps.
// f32 WMMA (V_WMMA_F32_16X16X4_F32) for the 200-row gate matvecs,
// weights in registers, state in LDS, 13 row-tile waves (416 threads).

#define Hh 50
#define Tt 65536
#define INn 8
#define G4 200     // 4*H gate rows
#define K1 64      // layer1 K padded: 8 inputs + 50 h + 6 zero
#define K2 100     // layer2 K: 50 h1 + 50 h2
#define NC1 16     // K1/4 WMMA chunks
#define NC2 25     // K2/4 WMMA chunks
#define NW 13      // row tiles / waves (208 rows >= 200)
#define NTHREADS (NW * 32)

typedef __attribute__((ext_vector_type(2))) float v2f;
typedef __attribute__((ext_vector_type(8))) float v8f;

__device__ __forceinline__ float sigm(float x) {
    return 1.0f / (1.0f + __expf(-x));
}

// Combined layer-1 weight row: [W_ih1 (8) | W_hh1 (50) | zeros (6)]
__device__ __forceinline__ float ldw1(const float* __restrict__ Wih,
                                      const float* __restrict__ Whh,
                                      int r, int k) {
    if (r >= G4) return 0.0f;
    if (k < INn) return Wih[r * INn + k];
    if (k < INn + Hh) return Whh[r * Hh + (k - INn)];
    return 0.0f;
}

// Combined layer-2 weight row: [W_ih2 (50) | W_hh2 (50)]
__device__ __forceinline__ float ldw2(const float* __restrict__ Wih,
                                      const float* __restrict__ Whh,
                                      int r, int k) {
    if (r >= G4) return 0.0f;
    if (k < Hh) return Wih[r * Hh + k];
    return Whh[r * Hh + (k - Hh)];
}

__global__ __launch_bounds__(NTHREADS, 1)
void lstm_persistent_kernel(const float* __restrict__ input,   // [T,8]
                            const float* __restrict__ Wih1,    // [200,8]
                            const float* __restrict__ Whh1,    // [200,50]
                            const float* __restrict__ bih1,    // [200]
                            const float* __restrict__ bhh1,    // [200]
                            const float* __restrict__ Wih2,    // [200,50]
                            const float* __restrict__ Whh2,    // [200,50]
                            const float* __restrict__ bih2,    // [200]
                            const float* __restrict__ bhh2,    // [200]
                            const float* __restrict__ Wout,    // [1,50]
                            const float* __restrict__ bout,    // [1]
                            float* __restrict__ out)           // [T]
{
    __shared__ float sv1[K1];      // [x0..x6, err, h1(50), pad(6)]
    __shared__ float sv2[K2];      // [h1(50), h2(50)]
    __shared__ float g1[NW * 16];  // raw layer-1 gates (208, 200 used)
    __shared__ float g2[NW * 16];  // raw layer-2 gates
    __shared__ float bsum1[G4];
    __shared__ float bsum2[G4];
    __shared__ float c1buf[Hh];
    __shared__ float c2buf[Hh];
    __shared__ float swout[Hh];
    __shared__ float sx7;          // current step's x_t[7] (act_dist)

    const int tid   = threadIdx.x;
    const int lane  = tid & 31;
    const int w     = tid >> 5;                 // wave / row-tile id 0..12
    const int mrow  = (w << 4) + (lane & 15);   // gate row this lane holds in A
    const int khalf = (lane >> 4) << 1;         // 0 (lanes 0-15) or 2 (16-31)

    // ---- Preload weight tiles into registers (read once, reused 65536x) ----
    v2f A1[NC1];
    v2f A2[NC2];
#pragma unroll
    for (int c = 0; c < NC1; ++c) {
        const int k0 = 4 * c + khalf;
        A1[c].x = ldw1(Wih1, Whh1, mrow, k0);
        A1[c].y = ldw1(Wih1, Whh1, mrow, k0 + 1);
    }
#pragma unroll
    for (int c = 0; c < NC2; ++c) {
        const int k0 = 4 * c + khalf;
        A2[c].x = ldw2(Wih2, Whh2, mrow, k0);
        A2[c].y = ldw2(Wih2, Whh2, mrow, k0 + 1);
    }
    const float bo = bout[0];

    // ---- Init LDS state (h=c=err=0, x row 0 installed) ----
    for (int j = tid; j < G4; j += NTHREADS) {
        bsum1[j] = bih1[j] + bhh1[j];
        bsum2[j] = bih2[j] + bhh2[j];
    }
    for (int j = tid; j < Hh; j += NTHREADS) {
        c1buf[j] = 0.0f;
        c2buf[j] = 0.0f;
        swout[j] = Wout[j];
    }
    for (int j = tid; j < K2; j += NTHREADS) sv2[j] = 0.0f;
    if (tid < 7)  sv1[tid] = input[tid];
    if (tid == 7) { sv1[7] = 0.0f; sx7 = input[7]; }
    if (tid >= 8 && tid < K1) sv1[tid] = 0.0f;
    __syncthreads();

    float err = 0.0f;  // live only in wave0 lane0

    for (int t = 0; t < Tt; ++t) {
        // Prefetch next input row early; consumed at end of step.
        float xn = 0.0f;
        if (tid < INn) {
            const int tn = (t + 1 < Tt) ? (t + 1) : t;
            xn = input[tn * INn + tid];
        }

        // ---- Phase A: gates1 = Wc1 @ [x, err, h1] via f32 WMMA ----
        {
            v8f acc = {};
#pragma unroll
            for (int c = 0; c < NC1; ++c) {
                const int k0 = 4 * c + khalf;
                v2f b;
                b.x = sv1[k0];
                b.y = sv1[k0 + 1];
                acc = __builtin_amdgcn_wmma_f32_16x16x4_f32(
                    false, A1[c], false, b, (short)0, acc, false, false);
            }
            if ((lane & 15) == 0) {  // lane 0 -> rows 0..7, lane 16 -> rows 8..15
                const int base = (w << 4) + ((lane >> 4) << 3);
#pragma unroll
                for (int i = 0; i < 8; ++i) g1[base + i] = acc[i];
            }
        }
        __syncthreads();

        // ---- Phase B: layer-1 activations (wave 0) ----
        if (tid < 32) {
            for (int j = lane; j < Hh; j += 32) {
                const float gi = sigm(g1[j]          + bsum1[j]);
                const float gf = sigm(g1[j + Hh]     + bsum1[j + Hh]);
                const float gg = tanhf(g1[j + 2*Hh]  + bsum1[j + 2*Hh]);
                const float go = sigm(g1[j + 3*Hh]   + bsum1[j + 3*Hh]);
                const float cn = gf * c1buf[j] + gi * gg;
                c1buf[j] = cn;
                const float hn = go * tanhf(cn);
                sv1[INn + j] = hn;   // next step's layer-1 input
                sv2[j]       = hn;   // layer-2 input (this step)
            }
        }
        __syncthreads();

        // ---- Phase C: gates2 = Wc2 @ [h1, h2] via f32 WMMA ----
        {
            v8f acc = {};
#pragma unroll
            for (int c = 0; c < NC2; ++c) {
                const int k0 = 4 * c + khalf;
                v2f b;
                b.x = sv2[k0];
                b.y = sv2[k0 + 1];
                acc = __builtin_amdgcn_wmma_f32_16x16x4_f32(
                    false, A2[c], false, b, (short)0, acc, false, false);
            }
            if ((lane & 15) == 0) {
                const int base = (w << 4) + ((lane >> 4) << 3);
#pragma unroll
                for (int i = 0; i < 8; ++i) g2[base + i] = acc[i];
            }
        }
        __syncthreads();

        // ---- Phase D: layer-2 activations, y = Wout.h2 + b, err update (wave 0) ----
        if (tid < 32) {
            float h2a, h2b = 0.0f;
            {
                const int j = lane;  // 0..31 < 50
                const float gi = sigm(g2[j]          + bsum2[j]);
                const float gf = sigm(g2[j + Hh]     + bsum2[j + Hh]);
                const float gg = tanhf(g2[j + 2*Hh]  + bsum2[j + 2*Hh]);
                const float go = sigm(g2[j + 3*Hh]   + bsum2[j + 3*Hh]);
                const float cn = gf * c2buf[j] + gi * gg;
                c2buf[j] = cn;
                h2a = go * tanhf(cn);
                sv2[Hh + j] = h2a;   // next step's layer-2 recurrent input
            }
            if (lane + 32 < Hh) {
                const int j = lane + 32;  // 32..49
                const float gi = sigm(g2[j]          + bsum2[j]);
                const float gf = sigm(g2[j + Hh]     + bsum2[j + Hh]);
                const float gg = tanhf(g2[j + 2*Hh]  + bsum2[j + 2*Hh]);
                const float go = sigm(g2[j + 3*Hh]   + bsum2[j + 3*Hh]);
                const float cn = gf * c2buf[j] + gi * gg;
                c2buf[j] = cn;
                h2b = go * tanhf(cn);
                sv2[Hh + j] = h2b;
            }
            // y = dot(Wout, h2) via lane-parallel partials + xor-shuffle reduce
            float p = swout[lane] * h2a;
            if (lane + 32 < Hh) p += swout[lane + 32] * h2b;
#pragma unroll
            for (int off = 16; off > 0; off >>= 1) p += __shfl_xor(p, off, 32);
            if (lane == 0) {
                const float y = p + bo;
                out[t] = y;
                err = 0.9f * err + 0.1f * (sx7 - y);
                sv1[7] = err;        // feedback input for next step
            }
            // install next x row
            if (lane < 7)  sv1[lane] = xn;
            if (lane == 7) sx7 = xn;
        }
        __syncthreads();
    }
}

extern "C" void kernel_launch(void* const* d_in, const int* in_sizes, int n_in,
                              void* d_out, int out_size, void* d_ws, size_t ws_size,
                              hipStream_t stream) {
    (void)in_sizes; (void)n_in; (void)d_ws; (void)ws_size; (void)out_size;
    const float* input = (const float*)d_in[0];
    const float* Wih1  = (const float*)d_in[1];
    const float* Whh1  = (const float*)d_in[2];
    const float* bih1  = (const float*)d_in[3];
    const float* bhh1  = (const float*)d_in[4];
    const float* Wih2  = (const float*)d_in[5];
    const float* Whh2  = (const float*)d_in[6];
    const float* bih2  = (const float*)d_in[7];
    const float* bhh2  = (const float*)d_in[8];
    const float* Wout  = (const float*)d_in[9];
    const float* bo    = (const float*)d_in[10];
    float* out = (float*)d_out;

    lstm_persistent_kernel<<<1, NTHREADS, 0, stream>>>(
        input, Wih1, Whh1, bih1, bhh1, Wih2, Whh2, bih2, bhh2, Wout, bo, out);
}